// ROIAlign_38001870635747
// MI455X (gfx1250) — compile-verified
//
#include <hip/hip_runtime.h>

// CDNA5 / gfx1250 ROIAlign using V_WMMA_F32_16X16X4_F32 for the
// y-interpolation + pooling stage (fp32 end to end; op is bandwidth bound).

typedef __attribute__((ext_vector_type(2))) float v2f;
typedef __attribute__((ext_vector_type(8))) float v8f;

// Only check the builtin during the device (gfx1250) pass; on the host pass
// amdgcn builtins are naturally absent.
#if defined(__HIP_DEVICE_COMPILE__) && defined(__AMDGCN__)
#if !__has_builtin(__builtin_amdgcn_wmma_f32_16x16x4_f32)
#error "gfx1250 f32 WMMA builtin __builtin_amdgcn_wmma_f32_16x16x4_f32 not available"
#endif
#endif

#define C_TOT  256
#define H_F    200
#define W_F    272
#define OUTSZ  7
#define NSAMP  14              // OUT * SR
#define NCOL   28              // 2 * NSAMP (lo/hi corner per sample)
#define CB     32              // channels per workgroup
#define SCALE_F 0.0625f

__global__ __launch_bounds__(256) void roialign_wmma_kernel(
    const float* __restrict__ feats,   // (N, 256, 200, 272) f32
    const float* __restrict__ rois,    // (K, 5) f32
    float* __restrict__ out)           // (K, 256, 7, 7) f32
{
    __shared__ float Qs[OUTSZ * CB * NCOL]; // [px][c][i], i contiguous (24.5 KB)
    __shared__ int   xcols[32];             // clamped x index per column j (28 used, padded)
    __shared__ float wxcs[32];              // 0.5 * x-weight * x-valid (padded with 0)
    __shared__ int   yrowW[NCOL];           // clamped y index * W per row i
    __shared__ float wys[NCOL];             // 0.5 * y-weight * y-valid

    const int tid  = threadIdx.x;
    const int lane = tid & 31;
    const int wv   = tid >> 5;              // wave id 0..7
    const int k    = blockIdx.y;            // ROI
    const int c0   = blockIdx.x * CB;       // channel block base

    const float* roi = rois + (size_t)k * 5;

    // ---- per-axis sample setup (matches reference axis()) ----
    if (tid < 2 * NSAMP) {
        const bool isY = (tid >= NSAMP);
        const int  s   = isY ? tid - NSAMP : tid;
        const float p0 = roi[isY ? 2 : 1] * SCALE_F;
        const float p1 = roi[isY ? 4 : 3] * SCALE_F;
        const float sz = fmaxf(p1 - p0, 1.0f);
        // offs = (s + 0.5)/SR ; coord = p0 + (sz/OUT)*offs
        const float coord = p0 + (sz * (1.0f / OUTSZ)) * ((s + 0.5f) * 0.5f);
        const int   dim   = isY ? H_F : W_F;
        const bool  valid = (coord > -1.0f) && (coord < (float)dim);
        const float cc    = fminf(fmaxf(coord, 0.0f), (float)(dim - 1));
        int   lo = (int)floorf(cc);
        int   hi = lo + 1; if (hi > dim - 1) hi = dim - 1;
        const float fr = cc - (float)lo;
        const float w0 = valid ? 0.5f * (1.0f - fr) : 0.0f; // 0.5 = half of the 1/4 pool mean
        const float w1 = valid ? 0.5f * fr          : 0.0f;
        if (isY) {
            yrowW[2 * s]     = lo * W_F;  wys[2 * s]     = w0;
            yrowW[2 * s + 1] = hi * W_F;  wys[2 * s + 1] = w1;
        } else {
            xcols[2 * s]     = lo;        wxcs[2 * s]     = w0;
            xcols[2 * s + 1] = hi;        wxcs[2 * s + 1] = w1;
        }
    } else if (tid < 32) {
        xcols[tid] = 0; wxcs[tid] = 0.0f;   // pad lanes 28..31
    }
    __syncthreads();

    // ---- stage 1: gather + x-interp + 2x-in-x pooling via lane butterfly ----
    // lane j = x-column (j = 2*sx + h); j>>2 == px (bin), j&3 == corner in bin.
    const int    b  = (int)roi[0];
    const float* fb = feats + ((size_t)b * C_TOT + c0) * (size_t)(H_F * W_F);
    const int    j  = lane;
    const int    xc = xcols[j];
    const float  wj = wxcs[j];

    for (int c = 0; c < CB; ++c) {
        const float* fc = fb + (size_t)c * (H_F * W_F);
        for (int ii = wv; ii < NCOL; ii += 8) {
            float v = 0.0f;
            if (j < NCOL) v = wj * fc[yrowW[ii] + xc];
            v += __shfl_xor(v, 1, 32);      // sum the 4 corner lanes of a bin
            v += __shfl_xor(v, 2, 32);
            if (((j & 3) == 0) && (j < NCOL))
                Qs[(((j >> 2) * CB) + c) * NCOL + ii] = v;
        }
    }
    __syncthreads();

    // ---- stage 2: out[py,px,c] = sum_i wy[i] * Q[i,px,c] via f32 WMMA ----
    // A (16x4 f32): lanes 0-15 hold M; VGPR0 = K0|K2, VGPR1 = K1|K3.
    // K-step kk covers i = 4kk..4kk+3, all of which belong to py == kk,
    // so A is nonzero only in row m == kk.
    const int m   = lane & 15;
    const int hi2 = (lane >> 4) << 1;       // 0 for lanes 0-15, 2 for lanes 16-31
    v2f afr[7];
#pragma unroll
    for (int kk = 0; kk < 7; ++kk) {
        float ax = 0.0f, ay = 0.0f;
        if (m == kk) { ax = wys[4 * kk + hi2]; ay = wys[4 * kk + hi2 + 1]; }
        afr[kk].x = ax; afr[kk].y = ay;
    }

    for (int T = wv; T < 14; T += 8) {      // 14 N-tiles = 7 px * 2 channel halves
        const int px = T >> 1;
        const int cg = (T & 1) << 4;
        v8f acc = {};
#pragma unroll
        for (int kk = 0; kk < 7; ++kk) {
            // B (4x16 f32): lane n = m, rows K0|K2 in .x, K1|K3 in .y
            const float* qp = &Qs[((px * CB) + cg + m) * NCOL + 4 * kk + hi2];
            v2f bfr; bfr.x = qp[0]; bfr.y = qp[1];
            acc = __builtin_amdgcn_wmma_f32_16x16x4_f32(
                false, afr[kk], false, bfr, (short)0, acc, false, false);
        }
        // D: VGPR r -> (M=r, N=lane) for lanes 0-15; rows 7..15 are padding.
        if (lane < 16) {
            float* op = out + ((size_t)k * C_TOT + (size_t)(c0 + cg + lane)) * (OUTSZ * OUTSZ) + px;
#pragma unroll
            for (int r = 0; r < 7; ++r) op[r * OUTSZ] = acc[r];
        }
    }
}

extern "C" void kernel_launch(void* const* d_in, const int* in_sizes, int n_in,
                              void* d_out, int out_size, void* d_ws, size_t ws_size,
                              hipStream_t stream) {
    (void)n_in; (void)out_size; (void)d_ws; (void)ws_size;
    const float* feats = (const float*)d_in[0];
    const float* rois  = (const float*)d_in[1];
    float*       out   = (float*)d_out;
    const int K = in_sizes[1] / 5;
    dim3 grid(C_TOT / CB, K);               // 8 channel blocks x K ROIs
    roialign_wmma_kernel<<<grid, dim3(256), 0, stream>>>(feats, rois, out);
}